// STFT_69982197121744
// MI455X (gfx1250) — compile-verified
//
#include <hip/hip_runtime.h>

typedef float v2f __attribute__((ext_vector_type(2)));
typedef float v8f __attribute__((ext_vector_type(8)));

namespace {
constexpr int kNFFT   = 1024;
constexpr int kHop    = 256;
constexpr int kBins   = 513;          // n_fft/2 + 1
constexpr int kCh     = 2 * kBins;    // 1026 output rows (cos || sin)
constexpr int kFrames = 1025;         // (262144 + 1024 - 1024)/256 + 1
constexpr int kXLen   = 262144;
constexpr int kBatch  = 16;
constexpr int kPadL   = 512;          // 'center' padding

constexpr int kMB   = 32;                         // channels per workgroup (2 m-tiles)
constexpr int kNB   = 128;                        // frames per workgroup (8 waves x 16)
constexpr int kSpan = (kNB - 1) * kHop + kNFFT;   // 33536 floats of padded signal
constexpr int kXPad = kSpan + 4 * (kSpan / 256);  // +4 floats pad per 256 -> bank stride 260
constexpr int kARow = 65;                         // float2 entries per k-step (64 + 1 pad)
constexpr int kA2Ent = 256 * kARow;               // 16640 float2 (swizzled, padded A)
// LDS layout: [0, kXPad) = signal window (small offsets -> DS immediates for B),
//             [kXPad, ..) = swizzled A tile (large base lives once in a register).
constexpr size_t kShmemBytes = (size_t)(kXPad + 2 * kA2Ent) * sizeof(float); // 269360 B
}

__global__ __launch_bounds__(256, 1)
void stft_wmma_f32(const float* __restrict__ x,
                   const float* __restrict__ dft,
                   float* __restrict__ out)
{
    extern __shared__ float smem[];
    float*  xb = smem;                                        // padded raw signal window
    float2* a2 = reinterpret_cast<float2*>(smem + kXPad);     // [256 k-steps][65 entries]

    const int t  = threadIdx.x;
    const int m0 = blockIdx.y * kMB;
    const int f0 = blockIdx.x * kNB;
    const int b  = blockIdx.z;

    // ---- Stage A: 32 rows x 1024 taps of the DFT basis, pre-swizzled into the
    // 16x16x4 A-operand lane layout. Thread t owns k-step row kb=t (cols 4t..4t+3)
    // for every basis row -> fully coalesced global float4 loads. The 1-entry row
    // pad (stride 520 B == 2 dwords mod 64 banks) makes fill stores (banks 2t+2w)
    // and compute loads (banks 2kb+2lane) conflict-free with zero swizzle math.
    {
        const int kb = t;
        const float* colp = dft + (kb << 2);
        float2* arow = a2 + kb * kARow;
        #pragma unroll 4
        for (int m = 0; m < kMB; ++m) {
            const int g = m0 + m;
            float4 v = make_float4(0.f, 0.f, 0.f, 0.f);
            if (g < kCh) v = *reinterpret_cast<const float4*>(colp + (size_t)g * kNFFT);
            const int w = ((m >> 4) << 5) | (m & 15);      // s*32 + m_local
            arow[w]      = make_float2(v.x, v.y);          // lanes 0-15: k,k+1
            arow[w + 16] = make_float2(v.z, v.w);          // lanes 16-31: k+2,k+3
        }
    }

    // ---- Stage B: contiguous padded-signal window for 128 overlapping frames,
    // zero-filled at the 'center' pad edges, 4-float pad every 256 for banks.
    {
        const float* xg = x + (size_t)b * kXLen;
        const int p0 = f0 * kHop;                  // start in padded coordinates
        for (int j = t; j < kSpan / 4; j += 256) {
            const int pos = j << 2;
            const int xi  = p0 + pos - kPadL;
            float4 v;
            if (xi >= 0 && xi + 3 < kXLen) {
                v = *reinterpret_cast<const float4*>(xg + xi);
            } else {
                v.x = (xi + 0 >= 0 && xi + 0 < kXLen) ? xg[xi + 0] : 0.f;
                v.y = (xi + 1 >= 0 && xi + 1 < kXLen) ? xg[xi + 1] : 0.f;
                v.z = (xi + 2 >= 0 && xi + 2 < kXLen) ? xg[xi + 2] : 0.f;
                v.w = (xi + 3 >= 0 && xi + 3 < kXLen) ? xg[xi + 3] : 0.f;
            }
            const int li = pos + ((pos >> 8) << 2);        // insert 4-float pad / 256
            *reinterpret_cast<float4*>(xb + li) = v;       // li % 4 == 0 -> aligned b128
        }
    }
    __syncthreads();

    // ---- Compute: each wave = one 16-frame strip x two 16-channel tiles,
    // 256 chained V_WMMA_F32_16X16X4_F32 per accumulator over K=1024.
    // All in-loop addresses are affine in the k-step: A advances 65 float2/step,
    // B advances 4 floats/step with one +4 pad hop per 64-step chunk, so both
    // reduce to a per-body vaddr bump + small DS immediate offsets.
    const int lane = t & 31;
    const int wv   = t >> 5;
    const int half = lane >> 4;                    // k-pair select (ISA A/B layout)
    const int lh   = lane & 15;

    const int fl     = wv * 16 + lh;               // frame within block
    const int bbase  = fl * kHop + 2 * half + 4 * fl;  // li at kb=0 (pad term folded)

    const float2* ap = a2 + lane;                  // += kARow per k-step
    const float*  bp = xb + bbase;                 // += 4 per k-step, += 4 per chunk

    v8f acc[2];
    acc[0] = (v8f){0.f, 0.f, 0.f, 0.f, 0.f, 0.f, 0.f, 0.f};
    acc[1] = acc[0];

    for (int c = 0; c < 4; ++c) {
        #pragma unroll 8
        for (int kk = 0; kk < 64; ++kk) {
            const v2f a0 = *reinterpret_cast<const v2f*>(ap);
            const v2f a1 = *reinterpret_cast<const v2f*>(ap + 32);
            const v2f bv = *reinterpret_cast<const v2f*>(bp);
            acc[0] = __builtin_amdgcn_wmma_f32_16x16x4_f32(false, a0, false, bv,
                                                           (short)0, acc[0], false, false);
            acc[1] = __builtin_amdgcn_wmma_f32_16x16x4_f32(false, a1, false, bv,
                                                           (short)0, acc[1], false, false);
            ap += kARow;
            bp += 4;
        }
        bp += 4;                                   // cross the 4-float LDS bank pad
    }

    // ---- Store: C/D layout -> VGPR r holds (M = r + 8*half, N = lh).
    // Row < 513 goes to the real half of d_out, else to the imag half.
    const int f = f0 + wv * 16 + lh;
    if (f < kFrames) {
        constexpr size_t kHalfOut = (size_t)kBatch * kBins * kFrames;
        #pragma unroll
        for (int s = 0; s < 2; ++s) {
            #pragma unroll
            for (int r = 0; r < 8; ++r) {
                const int m = m0 + s * 16 + r + 8 * half;
                if (m < kCh) {
                    const size_t o = (m < kBins)
                        ? (((size_t)b * kBins + m) * kFrames + f)
                        : (kHalfOut + ((size_t)b * kBins + (m - kBins)) * kFrames + f);
                    out[o] = acc[s][r];
                }
            }
        }
    }
}

extern "C" void kernel_launch(void* const* d_in, const int* in_sizes, int n_in,
                              void* d_out, int out_size, void* d_ws, size_t ws_size,
                              hipStream_t stream) {
    const float* x   = (const float*)d_in[0];   // (16, 1, 262144) f32
    const float* dft = (const float*)d_in[1];   // (1026, 1, 1024) f32
    float* out = (float*)d_out;                 // real(16,513,1025) || imag(16,513,1025)

    (void)in_sizes; (void)n_in; (void)out_size; (void)d_ws; (void)ws_size;

    // CDNA5 WGP has 320 KB LDS; we use 269 KB of dynamic LDS per workgroup.
    (void)hipFuncSetAttribute(reinterpret_cast<const void*>(stft_wmma_f32),
                              hipFuncAttributeMaxDynamicSharedMemorySize,
                              (int)kShmemBytes);

    dim3 grid((kFrames + kNB - 1) / kNB,   // 9 frame blocks
              (kCh + kMB - 1) / kMB,       // 33 channel blocks
              kBatch);                     // 16
    stft_wmma_f32<<<grid, 256, kShmemBytes, stream>>>(x, dft, out);
}